// GeometryFeatureExtractor_44727789420739
// MI455X (gfx1250) — compile-verified
//
#include <hip/hip_runtime.h>
#include <math.h>
#include <float.h>

typedef __attribute__((ext_vector_type(2))) float v2f;
typedef __attribute__((ext_vector_type(8))) float v8f;

#define BATCH 8
#define N_PTS 2048
#define DIM 512
#define ROWS_PER_BLK 32      // rows of the distance matrix per workgroup
#define COLS_PER_CHUNK 128   // columns processed per outer iteration (8 waves x 16)
#define K_CHUNK 64           // K slice staged in LDS for the B operand
#define LDA 516              // 512 + 4 pad floats  (516 % 64 == 4 -> conflict-free)
#define LDB 68               // 64 + 4 pad floats
#define LDD 132              // 128 + 4 pad floats, distance tile stride

// ---------------- pass 1: row squared norms ----------------
__global__ __launch_bounds__(256) void sqnorm_kernel(const float* __restrict__ x,
                                                     float* __restrict__ sq) {
    const int wave = threadIdx.x >> 5;
    const int lane = threadIdx.x & 31;
    const int row  = blockIdx.x * 8 + wave;          // 2048 blocks * 8 waves = 16384 rows
    const float* xr = x + (size_t)row * DIM;
    float s = 0.f;
    #pragma unroll 4
    for (int c = lane; c < DIM; c += 32) { float v = xr[c]; s += v * v; }
    #pragma unroll
    for (int off = 16; off >= 1; off >>= 1) s += __shfl_xor(s, off, 32);
    if (lane == 0) sq[row] = s;
}

__device__ inline void top5_insert(float v, float &a0, float &a1, float &a2,
                                   float &a3, float &a4) {
    if (v < a4) {
        a4 = v;
        if (a4 < a3) { float t = a3; a3 = a4; a4 = t;
            if (a3 < a2) { t = a2; a2 = a3; a3 = t;
                if (a2 < a1) { t = a1; a1 = a2; a2 = t;
                    if (a1 < a0) { t = a0; a0 = a1; a1 = t; } } } }
    }
}

// ---------------- pass 2: gram via fp32 WMMA + streaming top-5 / variance ----------------
__global__ __launch_bounds__(256) void knn_geom_kernel(const float* __restrict__ x,
                                                       const float* __restrict__ sq,
                                                       float* __restrict__ out) {
    __shared__ float ldsA[ROWS_PER_BLK * LDA];                 // 66048 B, resident A tile
    __shared__ float ldsB[COLS_PER_CHUNK * LDB];               // 34816 B, streamed B slice
    __shared__ float ldsD[ROWS_PER_BLK * LDD];                 // 16896 B, distance tile
    __shared__ float ldsM[ROWS_PER_BLK * 8 * 7];               //  7168 B, partial merge area
    __shared__ float ldsSqRow[ROWS_PER_BLK];

    const int t    = threadIdx.x;
    const int lane = t & 31;
    const int wave = t >> 5;                     // 0..7 -> column tile
    const int blk  = blockIdx.x;                 // 512 blocks
    const int b    = blk >> 6;                   // 64 row-blocks per batch
    const int rowBase = (blk & 63) * ROWS_PER_BLK;
    const float* xb  = x  + (size_t)b * N_PTS * DIM;
    const float* sqb = sq + b * N_PTS;

    // cooperative load of A tile (32 x 512) as float4
    {
        const int nVec = ROWS_PER_BLK * (DIM / 4);             // 4096 float4
        for (int idx = t; idx < nVec; idx += 256) {
            const int r  = idx >> 7;                           // 128 float4 per row
            const int c4 = idx & 127;
            float4 v = *(const float4*)&xb[(size_t)(rowBase + r) * DIM + c4 * 4];
            *(float4*)&ldsA[r * LDA + c4 * 4] = v;
        }
        if (t < ROWS_PER_BLK) ldsSqRow[t] = sqb[rowBase + t];
    }

    // per-thread scan state: thread owns row (t>>3), column-16-strip (t&7)
    const int sr = t >> 3;
    const int ss = t & 7;
    float n0 = FLT_MAX, n1 = FLT_MAX, n2 = FLT_MAX, n3 = FLT_MAX, n4 = FLT_MAX;
    float sumAll = 0.f, sumSqAll = 0.f;

    const int koff = (lane >> 4) * 2;            // f32 WMMA A/B lane K offset

    for (int cb = 0; cb < N_PTS / COLS_PER_CHUNK; ++cb) {
        const int colBase = cb * COLS_PER_CHUNK;
        v8f acc0 = {0.f, 0.f, 0.f, 0.f, 0.f, 0.f, 0.f, 0.f};  // rows 0..15
        v8f acc1 = {0.f, 0.f, 0.f, 0.f, 0.f, 0.f, 0.f, 0.f};  // rows 16..31

        for (int kb = 0; kb < DIM / K_CHUNK; ++kb) {
            // prefetch next K slice of the B panel: 256 cachelines, one per thread
            if (kb + 1 < DIM / K_CHUNK) {
                const float* pf = &xb[(size_t)(colBase + (t >> 1)) * DIM +
                                      (kb + 1) * K_CHUNK + (t & 1) * 32];
                __builtin_prefetch(pf, 0, 0);
            }
            __syncthreads();                     // previous readers of ldsB/ldsD done
            // stage B slice: 128 cols x 64 K as float4 (2048 float4 / 256 thr)
            #pragma unroll
            for (int i = 0; i < 8; ++i) {
                const int idx = t + i * 256;
                const int br  = idx >> 4;        // 16 float4 per row
                const int c4  = idx & 15;
                float4 v = *(const float4*)&xb[(size_t)(colBase + br) * DIM + kb * K_CHUNK + c4 * 4];
                *(float4*)&ldsB[br * LDB + c4 * 4] = v;
            }
            __syncthreads();

            const int aBase0 = (lane & 15) * LDA + kb * K_CHUNK;         // rows 0..15
            const int aBase1 = (16 + (lane & 15)) * LDA + kb * K_CHUNK;  // rows 16..31
            const int bBase  = (wave * 16 + (lane & 15)) * LDB;
            #pragma unroll
            for (int k = 0; k < K_CHUNK; k += 4) {
                v2f a0v, a1v, bv;
                a0v.x = ldsA[aBase0 + k + koff];
                a0v.y = ldsA[aBase0 + k + koff + 1];
                a1v.x = ldsA[aBase1 + k + koff];
                a1v.y = ldsA[aBase1 + k + koff + 1];
                bv.x  = ldsB[bBase + k + koff];
                bv.y  = ldsB[bBase + k + koff + 1];
                // two independent accumulation chains sharing the B fragment
                acc0 = __builtin_amdgcn_wmma_f32_16x16x4_f32(
                    false, a0v, false, bv, (short)0, acc0, false, false);
                acc1 = __builtin_amdgcn_wmma_f32_16x16x4_f32(
                    false, a1v, false, bv, (short)0, acc1, false, false);
            }
        }

        // gram tiles -> distance tile in LDS
        {
            const int cl = wave * 16 + (lane & 15);
            const float sqc = sqb[colBase + cl];
            const int rOff = (lane >> 4) << 3;   // lanes 16-31 hold rows +8
            #pragma unroll
            for (int j = 0; j < 8; ++j) {
                const int rl0 = j + rOff;
                float d2 = ldsSqRow[rl0] + sqc - 2.f * acc0[j];
                ldsD[rl0 * LDD + cl] = d2 > 0.f ? sqrtf(d2) : 0.f;
                const int rl1 = 16 + j + rOff;
                d2 = ldsSqRow[rl1] + sqc - 2.f * acc1[j];
                ldsD[rl1 * LDD + cl] = d2 > 0.f ? sqrtf(d2) : 0.f;
            }
        }
        __syncthreads();

        // streaming scan: 8 threads per row, 16 columns each
        #pragma unroll
        for (int c = 0; c < 16; ++c) {
            float v = ldsD[sr * LDD + ss * 16 + c];
            sumAll   += v;
            sumSqAll += v * v;
            top5_insert(v, n0, n1, n2, n3, n4);
        }
        // next loop's first __syncthreads() separates this read from the next ldsD write
    }

    // publish partials: 7 floats per (row, strip)
    {
        float* p = &ldsM[t * 7];
        p[0] = n0; p[1] = n1; p[2] = n2; p[3] = n3; p[4] = n4;
        p[5] = sumAll; p[6] = sumSqAll;
    }
    __syncthreads();

    if (t < ROWS_PER_BLK) {
        float m0 = FLT_MAX, m1 = FLT_MAX, m2 = FLT_MAX, m3 = FLT_MAX, m4 = FLT_MAX;
        float sA = 0.f, sqA = 0.f;
        #pragma unroll
        for (int p = 0; p < 8; ++p) {
            const float* q = &ldsM[(t * 8 + p) * 7];
            top5_insert(q[0], m0, m1, m2, m3, m4);
            top5_insert(q[1], m0, m1, m2, m3, m4);
            top5_insert(q[2], m0, m1, m2, m3, m4);
            top5_insert(q[3], m0, m1, m2, m3, m4);
            top5_insert(q[4], m0, m1, m2, m3, m4);
            sA  += q[5];
            sqA += q[6];
        }
        // tree_ness = d5 / max(d2, eps)
        const float tree = m4 / fmaxf(m1, 1e-6f);
        // cycle_ness = 1 / (var(neighbors, ddof=1) + eps)
        const float mean5 = (m0 + m1 + m2 + m3 + m4) * 0.2f;
        const float v5 = ((m0 - mean5) * (m0 - mean5) + (m1 - mean5) * (m1 - mean5) +
                          (m2 - mean5) * (m2 - mean5) + (m3 - mean5) * (m3 - mean5) +
                          (m4 - mean5) * (m4 - mean5)) * (1.f / 4.f);
        const float cyc = 1.f / (v5 + 1e-6f);
        // flat_ness = 1 / (var(all dists, ddof=1) + eps)
        const float meanA = sA * (1.f / (float)N_PTS);
        const float varA  = (sqA - sA * meanA) * (1.f / (float)(N_PTS - 1));
        const float flat  = 1.f / (varA + 1e-6f);

        const size_t o = ((size_t)b * N_PTS + rowBase + t) * 3;
        out[o + 0] = 1.f / (1.f + expf(-tree * 0.1f));
        out[o + 1] = 1.f / (1.f + expf(-cyc  * 0.1f));
        out[o + 2] = 1.f / (1.f + expf(-flat * 0.1f));
    }
}

extern "C" void kernel_launch(void* const* d_in, const int* in_sizes, int n_in,
                              void* d_out, int out_size, void* d_ws, size_t ws_size,
                              hipStream_t stream) {
    (void)in_sizes; (void)n_in; (void)out_size; (void)ws_size;
    const float* x  = (const float*)d_in[0];
    float* out      = (float*)d_out;
    float* sq       = (float*)d_ws;              // 16384 floats = 64 KB

    sqnorm_kernel<<<(BATCH * N_PTS) / 8, 256, 0, stream>>>(x, sq);
    knn_geom_kernel<<<(BATCH * N_PTS) / ROWS_PER_BLK, 256, 0, stream>>>(x, sq, out);
}